// GCN_94489281062
// MI455X (gfx1250) — compile-verified
//
#include <hip/hip_runtime.h>
#include <hip/hip_bf16.h>

// GCN 3-layer forward for MI455X (gfx1250, wave32).
// GEMMs: V_WMMA_F32_16X16X4_F32, all shapes compile-time (no predication).
// Layer-3 weight is zero-padded 128x40 -> 128x48 so every tile is full.
// Edge scatter: wave-per-edge fp32 atomics, L2-resident working set.

#define NN      100000
#define EE      1600000
#define IN_DIM  256
#define HID     128
#define NCLS    40
#define NCLS_P  48      // padded to 3 full 16-col WMMA tiles

typedef __attribute__((ext_vector_type(2))) float v2f;
typedef __attribute__((ext_vector_type(8))) float v8f;

// ---------------------------------------------------------------------------
// degree / normalization
// ---------------------------------------------------------------------------
__global__ void init_deg_kernel(float* deg, int n) {
    int i = blockIdx.x * blockDim.x + threadIdx.x;
    if (i < n) deg[i] = 1.0f;  // self-loop contributes 1 to in-degree
}

__global__ void deg_accum_kernel(const int* __restrict__ dst, float* deg, int e) {
    int i = blockIdx.x * blockDim.x + threadIdx.x;
    int stride = gridDim.x * blockDim.x;
    for (; i < e; i += stride) atomicAdd(&deg[dst[i]], 1.0f);
}

__global__ void deg_to_dinv_kernel(float* deg, int n) {
    int i = blockIdx.x * blockDim.x + threadIdx.x;
    if (i < n) deg[i] = rsqrtf(deg[i]);  // deg >= 1 always (self-loops)
}

// Zero-pad W3 [HID x NCLS] into W3p [HID x NCLS_P]
__global__ void pad_w3_kernel(const float* __restrict__ w, float* __restrict__ wp) {
    int i = blockIdx.x * blockDim.x + threadIdx.x;           // over HID*NCLS_P
    if (i >= HID * NCLS_P) return;
    int r = i / NCLS_P, c = i - r * NCLS_P;
    wp[i] = (c < NCLS) ? w[r * NCLS + c] : 0.0f;
}

// ---------------------------------------------------------------------------
// GEMM: H[M x NOUT] = A[M x K] * B[K x NOUT], fp32 WMMA 16x16x4.
// One wave owns a 16-row x NOUT strip (NT = NOUT/16 accumulators); the A
// fragment (one float2/lane) is reused across all NT column tiles.
// A-frag (ISA 7.12.2, 32-bit 16x4): lane -> row = lane&15, K = (lane>>4)*2..+1
// B-frag (4x16):                    lane -> col = lane&15, K = (lane>>4)*2..+1
// D (16x16 f32): VGPR r -> M = r + 8*(lane>>4), N = lane&15
// ---------------------------------------------------------------------------
template <int NT, int K, int NOUT>
__global__ void gemm_wmma_f32_kernel(const float* __restrict__ A,
                                     const float* __restrict__ B,
                                     float* __restrict__ H,
                                     int M) {
    static_assert(NOUT == NT * 16, "full tiles only");
    const int lane  = threadIdx.x & 31;
    const int wave  = threadIdx.x >> 5;
    const int wpb   = blockDim.x >> 5;
    const int tile  = blockIdx.x * wpb + wave;     // 16-row tile index
    const int mtile = M >> 4;
    if (tile >= mtile) return;

    const int rowA    = (tile << 4) + (lane & 15);
    const int kb      = (lane >> 4) << 1;          // 0 or 2
    const int colBase = lane & 15;

    v8f acc[NT];
#pragma unroll
    for (int t = 0; t < NT; ++t) acc[t] = (v8f)0.0f;

    const float* __restrict__ arow = A + (size_t)rowA * K;

#pragma unroll 2
    for (int k = 0; k < K; k += 4) {
        const float2 av = *(const float2*)(arow + k + kb);   // one b64 load
        v2f a; a.x = av.x; a.y = av.y;

        const float* __restrict__ b0 = B + (size_t)(k + kb) * NOUT + colBase;
        const float* __restrict__ b1 = b0 + NOUT;
#pragma unroll
        for (int t = 0; t < NT; ++t) {
            v2f b;
            b.x = b0[t * 16];
            b.y = b1[t * 16];
            acc[t] = __builtin_amdgcn_wmma_f32_16x16x4_f32(
                false, a, false, b, (short)0, acc[t], false, false);
        }
    }

    const int rowD = (tile << 4) + ((lane >> 4) << 3);
    float* __restrict__ hout = H + (size_t)rowD * NOUT + colBase;
#pragma unroll
    for (int t = 0; t < NT; ++t) {
#pragma unroll
        for (int r = 0; r < 8; ++r)
            hout[(size_t)r * NOUT + t * 16] = acc[t][r];
    }
}

// ---------------------------------------------------------------------------
// Edge scatter: out[dst][0..F) += h[src][0..F) * dinv[src]*dinv[dst].
// One wave per edge; h has row stride HS (HS == F except padded layer 3).
// ---------------------------------------------------------------------------
template <int F, int HS>
__global__ void scatter_kernel(const float* __restrict__ h,
                               const int* __restrict__ src,
                               const int* __restrict__ dst,
                               const float* __restrict__ dinv,
                               float* __restrict__ out,
                               int e) {
    const int lane = threadIdx.x & 31;
    int warp = (blockIdx.x * blockDim.x + threadIdx.x) >> 5;
    const int nwarps = (gridDim.x * blockDim.x) >> 5;
    for (; warp < e; warp += nwarps) {
        const int s = src[warp];
        const int d = dst[warp];
        const float w = dinv[s] * dinv[d];
        const float* __restrict__ hs = h + (size_t)s * HS;
        float* __restrict__ od       = out + (size_t)d * F;
#pragma unroll
        for (int f = lane; f < F; f += 32)
            atomicAdd(&od[f], hs[f] * w);
    }
}

// ---------------------------------------------------------------------------
// out[i][f] = [relu]( agg[i][f] + h[i][f]*dinv[i]^2 + bias[f] )
// agg/out have row stride F; h has row stride HS.
// ---------------------------------------------------------------------------
template <int F, int HS, int RELU>
__global__ void finalize_kernel(const float* __restrict__ agg,
                                const float* __restrict__ h,
                                const float* __restrict__ dinv,
                                const float* __restrict__ bias,
                                float* __restrict__ out,
                                int n) {
    size_t idx = (size_t)blockIdx.x * blockDim.x + threadIdx.x;
    const size_t total = (size_t)n * F;
    const size_t stride = (size_t)gridDim.x * blockDim.x;
    for (; idx < total; idx += stride) {
        const int i = (int)(idx / F);           // F compile-time: shift/magic
        const int f = (int)(idx - (size_t)i * F);
        const float di = dinv[i];
        float v = agg[idx] + h[(size_t)i * HS + f] * di * di + bias[f];
        if (RELU) v = fmaxf(v, 0.0f);
        out[idx] = v;
    }
}

// ---------------------------------------------------------------------------
extern "C" void kernel_launch(void* const* d_in, const int* in_sizes, int n_in,
                              void* d_out, int out_size, void* d_ws, size_t ws_size,
                              hipStream_t stream) {
    const float* x    = (const float*)d_in[0];
    const int*   eidx = (const int*)d_in[1];
    const float* W1   = (const float*)d_in[2];
    const float* b1   = (const float*)d_in[3];
    const float* W2   = (const float*)d_in[4];
    const float* b2   = (const float*)d_in[5];
    const float* W3   = (const float*)d_in[6];
    const float* b3   = (const float*)d_in[7];
    float* out = (float*)d_out;

    const int* src = eidx;        // edge_index[0]
    const int* dst = eidx + EE;   // edge_index[1]

    // workspace (floats): dinv[N] | bufA[N*HID] | bufB[N*HID] | bufC[N*HID] | W3p[HID*48]
    float* dinv = (float*)d_ws;
    float* bufA = dinv + NN;                       // h (per-layer dense output)
    float* bufB = bufA + (size_t)NN * HID;         // agg1 -> x2 -> x3
    float* bufC = bufB + (size_t)NN * HID;         // agg2
    float* W3p  = bufC + (size_t)NN * HID;         // padded layer-3 weight

    const int TB = 256;

    // ---- normalization: deg -> dinv (in place) ----
    init_deg_kernel<<<(NN + TB - 1) / TB, TB, 0, stream>>>(dinv, NN);
    deg_accum_kernel<<<4096, TB, 0, stream>>>(dst, dinv, EE);
    deg_to_dinv_kernel<<<(NN + TB - 1) / TB, TB, 0, stream>>>(dinv, NN);
    pad_w3_kernel<<<(HID * NCLS_P + TB - 1) / TB, TB, 0, stream>>>(W3, W3p);

    const int MT = NN / 16;                 // 6250 row-tiles
    const int gemmBlocks = (MT + 7) / 8;    // 8 waves per 256-thread block
    const int scatterBlocks = 16384;        // grid-stride, 128k waves
    const int elemBlocks128 = (int)(((size_t)NN * HID + TB - 1) / TB);
    const int elemBlocks40  = (int)(((size_t)NN * NCLS + TB - 1) / TB);

    // ---- layer 1: h1 = x @ W1 ; agg ; x2 = relu(agg + h1*d^2 + b1) ----
    gemm_wmma_f32_kernel<8, IN_DIM, HID><<<gemmBlocks, TB, 0, stream>>>(x, W1, bufA, NN);
    hipMemsetAsync(bufB, 0, (size_t)NN * HID * sizeof(float), stream);
    scatter_kernel<HID, HID><<<scatterBlocks, TB, 0, stream>>>(bufA, src, dst, dinv, bufB, EE);
    finalize_kernel<HID, HID, 1><<<elemBlocks128, TB, 0, stream>>>(bufB, bufA, dinv, b1, bufB, NN);

    // ---- layer 2: h2 = x2 @ W2 ; agg ; x3 = relu(agg + h2*d^2 + b2) ----
    gemm_wmma_f32_kernel<8, HID, HID><<<gemmBlocks, TB, 0, stream>>>(bufB, W2, bufA, NN);
    hipMemsetAsync(bufC, 0, (size_t)NN * HID * sizeof(float), stream);
    scatter_kernel<HID, HID><<<scatterBlocks, TB, 0, stream>>>(bufA, src, dst, dinv, bufC, EE);
    finalize_kernel<HID, HID, 1><<<elemBlocks128, TB, 0, stream>>>(bufC, bufA, dinv, b2, bufB, NN);

    // ---- layer 3: h3 = x3 @ W3p (stride 48) ; agg into d_out ; out = agg + h3*d^2 + b3 ----
    gemm_wmma_f32_kernel<3, HID, NCLS_P><<<gemmBlocks, TB, 0, stream>>>(bufB, W3p, bufA, NN);
    hipMemsetAsync(out, 0, (size_t)NN * NCLS * sizeof(float), stream);
    scatter_kernel<NCLS, NCLS_P><<<scatterBlocks, TB, 0, stream>>>(bufA, src, dst, dinv, out, EE);
    finalize_kernel<NCLS, NCLS_P, 0><<<elemBlocks40, TB, 0, stream>>>(out, bufA, dinv, b3, out, NN);
}